// SparseMaskHead_41781441855751
// MI455X (gfx1250) — compile-verified
//
#include <hip/hip_runtime.h>
#include <hip/hip_bf16.h>
#include <stdint.h>

#define HH 400
#define WW 400
#define HWE (HH*WW)            // 160000
#define NP 20000
#define NANCH 2500
#define ROWW 16                // padded words per bitmask row (13 used)
#define NWORDS (HH*ROWW)       // 6400
#define TPB 1024
#define NCHUNK ((HWE + TPB - 1)/TPB)   // 157
#define WPT ((NWORDS + TPB - 1)/TPB)   // 7

typedef float v2f __attribute__((ext_vector_type(2)));
typedef float v8f __attribute__((ext_vector_type(8)));

// ---- CDNA5 async global->LDS copy path (ASYNCcnt) ----
__device__ __forceinline__ void async_ld_b32(unsigned lds_addr, const float* g) {
  asm volatile("global_load_async_to_lds_b32 %0, %1, off"
               :: "v"(lds_addr), "v"((unsigned long long)(uintptr_t)g)
               : "memory");
}
__device__ __forceinline__ void wait_async_le1() { asm volatile("s_wait_asynccnt 0x1" ::: "memory"); }
__device__ __forceinline__ void wait_async_le0() { asm volatile("s_wait_asynccnt 0x0" ::: "memory"); }

__global__ __launch_bounds__(TPB)
void sparse_mask_head_kernel(const float* __restrict__ pred,
                             const float* __restrict__ pmask,
                             const float* __restrict__ grid,
                             const float* __restrict__ zs,
                             float* __restrict__ out, int B) {
  // dilB doubles as the 4096-bin histogram during phase 1 (keeps LDS < 64KB).
  __shared__ unsigned anchorB[NWORDS];     // 25.6 KB
  __shared__ unsigned dilB[NWORDS];        // 25.6 KB (hist overlay)
  __shared__ unsigned scanBuf[TPB];        // 4 KB
  __shared__ unsigned waveSum[32];
  __shared__ unsigned rowTotS[64];         // 4 blocks x 16 row totals
  __shared__ unsigned blockTotS[4];
  __shared__ float    predStage[2][TPB];   // 8 KB async staging
  __shared__ unsigned selBinS, kRemS, tieBaseS, chunkTotS;

  unsigned* hist = dilB;

  const int tid = threadIdx.x;
  const int b   = blockIdx.x;
  const size_t base = (size_t)b * HWE;

  // output sections (all float; ints are value-cast)
  float* outIdx = out;                               // B*NP
  float* qpos   = out + (size_t)B * NP;              // B*4*NP*3
  float* nmask  = qpos + (size_t)B * 4 * NP * 3;     // B*HWE
  float* xyv    = nmask + (size_t)B * HWE;           // B*NP*2
  float* ign    = xyv + (size_t)B * NP * 2;          // B*(HWE-NP)

  float zsr[4];
  zsr[0]=zs[0]; zsr[1]=zs[1]; zsr[2]=zs[2]; zsr[3]=zs[3];

  // ---- Phase 1: exact top-NANCH threshold key via 12/12/8-bit radix refine ----
  // score = sigmoid(pred)*mask >= 0, so raw float bits are an order-preserving key.
  if (tid == 0) kRemS = NANCH;
  unsigned pref = 0;
  for (int level = 0; level < 3; ++level) {
    for (int j = tid; j < 4096; j += TPB) hist[j] = 0u;
    __syncthreads();
    for (int i = tid; i < HWE; i += TPB) {
      __builtin_prefetch(&pred[base + i + 4*TPB], 0, 1);   // -> global_prefetch_b8
      float sc = pmask[base+i] / (1.0f + __expf(-pred[base+i]));
      unsigned key = __float_as_uint(sc);
      bool cand; unsigned bin;
      if (level == 0)      { cand = true;                       bin = key >> 20; }
      else if (level == 1) { cand = (key >> 20) == (pref>>20);  bin = (key >> 8) & 0xFFFu; }
      else                 { cand = (key >> 8)  == (pref>>8);   bin = key & 0xFFu; }
      if (cand) atomicAdd(&hist[bin], 1u);                 // LDS ds_add
    }
    __syncthreads();
    if (tid == 0) {
      unsigned kRem = kRemS, acc = 0; int nb = (level==2)?256:4096; int bsel = 0;
      for (int bn = nb-1; bn >= 0; --bn) {
        acc += hist[bn];
        if (acc >= kRem) { bsel = bn; kRemS = kRem - (acc - hist[bn]); break; }
      }
      selBinS = (unsigned)bsel;
    }
    __syncthreads();
    pref |= selBinS << ((level==0) ? 20 : ((level==1) ? 8 : 0));
    __syncthreads();
  }
  const unsigned T = pref;           // threshold key (2500th largest)
  const unsigned tieQuota = kRemS;   // #elements equal to T to accept (lowest indices)

  // ---- Phase 2: build anchor bitmask with index-ordered tie-breaking ----
  for (int j = tid; j < NWORDS; j += TPB) anchorB[j] = 0u;
  if (tid == 0) tieBaseS = 0u;
  __syncthreads();

  // Every wave issues every chunk (addresses clamped) so the per-wave ASYNCcnt
  // waits below are always correct for the double buffer.
  async_ld_b32((unsigned)(uintptr_t)&predStage[0][tid], &pred[base + tid]);

  for (int c = 0; c < NCHUNK; ++c) {
    const int cur = c & 1, nxt = cur ^ 1;
    if (c + 1 < NCHUNK) {
      int i2 = (c+1)*TPB + tid;
      if (i2 >= HWE) i2 = HWE - 1;              // clamp: keep EXEC full
      async_ld_b32((unsigned)(uintptr_t)&predStage[nxt][tid], &pred[base + i2]);
      wait_async_le1();                         // oldest (current) chunk complete
    } else {
      wait_async_le0();
    }
    int i = c*TPB + tid;
    bool valid = i < HWE;
    bool isGt=false, isEq=false;
    if (valid) {
      float sc = pmask[base+i] / (1.0f + __expf(-predStage[cur][tid]));
      unsigned key = __float_as_uint(sc);
      isGt = key > T;
      isEq = key == T;
    }
    // wave32 ballot -> exact index-ordered rank among ties
    unsigned balu = (unsigned)__ballot(isEq);
    int lane = tid & 31, wv = tid >> 5;
    unsigned lanePre = __popc(balu & ((1u << lane) - 1u));
    if (lane == 0) waveSum[wv] = __popc(balu);
    __syncthreads();
    if (tid == 0) {
      unsigned a = 0;
      for (int w = 0; w < 32; ++w) { unsigned v = waveSum[w]; waveSum[w] = a; a += v; }
      chunkTotS = a;
    }
    __syncthreads();
    bool anchor = isGt || (isEq && (tieBaseS + waveSum[wv] + lanePre) < tieQuota);
    if (valid && anchor) {
      int r = i / WW, cc = i % WW;
      atomicOr(&anchorB[r*ROWW + (cc >> 5)], 1u << (cc & 31));
    }
    __syncthreads();
    if (tid == 0) tieBaseS += chunkTotS;
    __syncthreads();
  }

  // ---- Phase 3: 5x5 binary dilation directly on the bitmask ----
  for (int w = tid; w < NWORDS; w += TPB) {
    int r = w >> 4, wc = w & 15;
    unsigned d = 0u;
    if (wc < 13) {
      int r0 = (r >= 2) ? r-2 : 0;
      int r1 = (r <= HH-3) ? r+2 : HH-1;
      for (int rr = r0; rr <= r1; ++rr) {
        unsigned C = anchorB[rr*ROWW + wc];
        unsigned L = (wc > 0)  ? anchorB[rr*ROWW + wc - 1] : 0u;
        unsigned R = (wc < 12) ? anchorB[rr*ROWW + wc + 1] : 0u;
        d |= C | (C<<1) | (L>>31) | (C<<2) | (L>>30)
               | (C>>1) | (R<<31) | (C>>2) | (R<<30);
      }
      if (wc == 12) d &= 0xFFFFu;    // cols 384..399 only
    }
    dilB[w] = d;
  }
  __syncthreads();

  // ---- Phase 4: inclusive scan of 1024 popcount sums via WMMA tensor-core scan ----
  // Waves 0..3 each scan a 256-value block: S1 = D x U (U = upper-tri ones)
  // as 4 chained V_WMMA_F32_16X16X4_F32, then add row-carries from S1[.][15].
  const int w0 = tid * WPT;
  const int w1 = (w0 + WPT < NWORDS) ? (w0 + WPT) : NWORDS;
  unsigned local = 0;
  for (int w = w0; w < w1; ++w) local += __popc(dilB[w]);
  scanBuf[tid] = local;
  __syncthreads();

  v8f acc = {};
  const int lane = tid & 31;
  const int half = lane >> 4;        // 0: lanes 0-15, 1: lanes 16-31
  const int mrow = lane & 15;
  if ((tid >> 5) < 4) {              // wave-uniform: EXEC all-ones inside
    const int wblk = tid >> 5;
    const unsigned* x = &scanBuf[wblk * 256];
    #pragma unroll
    for (int kk = 0; kk < 4; ++kk) {
      // A[M][Kl] = D[M][4kk+Kl] = x[16*M + 4kk + Kl], Kl = vgpr + 2*half
      v2f A; A.x = (float)x[16*mrow + 4*kk + 2*half + 0];
             A.y = (float)x[16*mrow + 4*kk + 2*half + 1];
      // B[Kl][N] = U[4kk+Kl][N] = ((4kk+Kl) <= N), N = mrow
      v2f Bm; Bm.x = ((4*kk + 2*half + 0) <= mrow) ? 1.0f : 0.0f;
              Bm.y = ((4*kk + 2*half + 1) <= mrow) ? 1.0f : 0.0f;
      acc = __builtin_amdgcn_wmma_f32_16x16x4_f32(false, A, false, Bm,
                                                  (short)0, acc, false, false);
    }
    // acc[j] = S1[row = j + 8*half][col = mrow]; row totals live at col 15.
    if (mrow == 15) {
      #pragma unroll
      for (int j = 0; j < 8; ++j) rowTotS[wblk*16 + j + 8*half] = (unsigned)acc[j];
    }
  }
  __syncthreads();
  if ((tid >> 5) < 4) {
    const int wblk = tid >> 5;
    unsigned prefix[17]; prefix[0] = 0;
    #pragma unroll
    for (int r = 0; r < 16; ++r) prefix[r+1] = prefix[r] + rowTotS[wblk*16 + r];
    #pragma unroll
    for (int j = 0; j < 8; ++j) {
      int rj = j + 8*half;
      scanBuf[wblk*256 + 16*rj + mrow] = (unsigned)acc[j] + prefix[rj];
    }
    if (lane == 0) blockTotS[wblk] = prefix[16];
  }
  __syncthreads();

  const int blk = tid >> 8;
  unsigned boff = 0;
  #pragma unroll
  for (int q = 0; q < 4; ++q) if (q < blk) boff += blockTotS[q];
  const unsigned total = blockTotS[0] + blockTotS[1] + blockTotS[2] + blockTotS[3];
  unsigned ones = (scanBuf[tid] + boff) - local;   // exclusive prefix of ones

  // ---- Phase 5: emit all five outputs in one pass ----
  for (int w = w0; w < w1; ++w) {
    int r = w >> 4, wc = w & 15;
    if (wc >= 13) continue;
    unsigned m = dilB[w];
    int nb = (wc == 12) ? 16 : 32;
    int cb = wc * 32;
    for (int bb = 0; bb < nb; ++bb) {
      int i = r*WW + cb + bb;
      bool one = (m >> bb) & 1u;
      unsigned pos = one ? ones : (total + (unsigned)i - ones);
      if (one) ones++;
      bool sel = pos < NP;
      nmask[base + i] = sel ? 1.0f : 0.0f;
      if (sel) {
        size_t op = (size_t)b*NP + pos;
        outIdx[op]   = (float)i;
        xyv[op*2]    = (float)r;            // out_idx // H
        xyv[op*2+1]  = (float)(cb + bb);    // out_idx %  W
        float gx = grid[(size_t)i*2], gy = grid[(size_t)i*2 + 1];
        #pragma unroll
        for (int p = 0; p < 4; ++p) {
          size_t qb = (((size_t)b*4 + p)*NP + pos)*3;
          qpos[qb] = gx; qpos[qb+1] = gy; qpos[qb+2] = zsr[p];
        }
      } else {
        ign[(size_t)b*(HWE-NP) + (pos - NP)] = (float)i;
      }
    }
  }
}

extern "C" void kernel_launch(void* const* d_in, const int* in_sizes, int n_in,
                              void* d_out, int out_size, void* d_ws, size_t ws_size,
                              hipStream_t stream) {
  const float* pred  = (const float*)d_in[0];   // (B,1,400,400) f32
  const float* pmask = (const float*)d_in[1];   // (B,400,400)   f32
  const float* grid  = (const float*)d_in[2];   // (400,400,2)   f32
  const float* zs    = (const float*)d_in[3];   // (4,)          f32
  float* out = (float*)d_out;
  int B = in_sizes[0] / HWE;
  sparse_mask_head_kernel<<<dim3(B), dim3(TPB), 0, stream>>>(pred, pmask, grid, zs, out, B);
}